// ChamferLoss_18098992185717
// MI455X (gfx1250) — compile-verified
//
#include <hip/hip_runtime.h>
#include <hip/hip_bf16.h>
#include <math.h>

// CDNA5 / gfx1250 Chamfer distance using V_WMMA_F32_16X16X4_F32.
//
// Per 16x16 (n,m) tile:  D = A(16x4) x B(4x16) + 0
//   A row n  = (x0, x1, x2, 1)
//   B col m  = (-2*y0, -2*y1, -2*y2, ||y||^2)
//   => D[n][m] = ||y_m||^2 - 2 x_n . y_m        (xx[n] added after the min,
//                                                since it is constant per row)
// A lane layout (ISA 7.12.2, 32-bit 16x4): lanes 0-15 hold K=0 (V0), K=1 (V1);
// lanes 16-31 hold K=2 (V0), K=3 (V1). B (4x16) mirrors this per column.
// C/D (16x16): VGPR v, lanes 0-15 -> (M=v, N=lane); lanes 16-31 -> (M=v+8).
//
// Hot loop per tile: 1 ds_load_b64 (B fragment, bank-conflict-free float4
// layout) + 1 v_wmma with inline-0 C + elementwise min (fuses to v_min3).

typedef __attribute__((ext_vector_type(2))) float v2f;
typedef __attribute__((ext_vector_type(4))) float v4f;
typedef __attribute__((ext_vector_type(8))) float v8f;

#define CHUNK 1024          // search points staged in LDS per iteration
#define ROWS_PER_WAVE 16
#define WAVES_PER_BLOCK 8
#define ROWS_PER_BLOCK (ROWS_PER_WAVE * WAVES_PER_BLOCK)  // 128
#define BIGF 3.0e38f

// For each "row" point in x, find min squared distance to any point in y.
// One partial sum (over this wave's 16 rows) written per wave.
// Requires: Nx % 128 == 0, Ny % CHUNK == 0 (true for 8192).
__global__ void __launch_bounds__(256)
chamfer_nn_kernel(const float* __restrict__ x, const float* __restrict__ y,
                  int Nx, int Ny, int blocksPerBatch, float* __restrict__ partial)
{
    __shared__ v4f lpt[CHUNK];    // per point: (-2y0, -2y1, -2y2, ||y||^2)

    const int tid    = threadIdx.x;
    const int lane   = tid & 31;
    const int waveId = tid >> 5;
    const int col    = lane & 15;     // column (or row) within the 16-wide tile
    const int halfg  = lane >> 4;     // 0: K=0,1   1: K=2,3

    const int b       = blockIdx.x / blocksPerBatch;
    const int rowBase = (blockIdx.x % blocksPerBatch) * ROWS_PER_BLOCK
                        + waveId * ROWS_PER_WAVE;

    // ---- A fragment (constant across the whole m-loop) ----
    const int n = rowBase + col;
    const float* xp = x + ((size_t)b * Nx + n) * 3;
    const float x0 = xp[0], x1 = xp[1], x2 = xp[2];
    const float xx = x0 * x0 + x1 * x1 + x2 * x2;
    v2f afrag;
    afrag[0] = halfg ? x2 : x0;       // K=2 : K=0
    afrag[1] = halfg ? 1.0f : x1;     // K=3 : K=1   (A[n][3] = 1)

    // Running column-wise minima of (yy - 2 x.y) in D-fragment layout.
    v8f minD;
    #pragma unroll
    for (int i = 0; i < 8; ++i) minD[i] = BIGF;

    const float* ybase = y + (size_t)b * Ny * 3;
    const float* lptf  = (const float*)lpt;

    for (int chunk = 0; chunk < Ny; chunk += CHUNK) {
        __syncthreads();
        // Stage CHUNK search points, pre-transformed to B-fragment form.
        #pragma unroll
        for (int p = tid; p < CHUNK; p += 256) {
            const float* yp = ybase + (size_t)(chunk + p) * 3;
            const float a = yp[0], bb = yp[1], c = yp[2];
            v4f e;
            e[0] = -2.0f * a;
            e[1] = -2.0f * bb;
            e[2] = -2.0f * c;
            e[3] = a * a + bb * bb + c * c;
            lpt[p] = e;
        }
        __syncthreads();

        // 64 tiles of 16 columns: 1 b64 LDS load + 1 WMMA (+ fused min3) each.
        #pragma unroll 4
        for (int t = 0; t < CHUNK; t += 16) {
            // byte addr = 16*(t+col) + 8*halfg  -> 8-aligned, conflict-free
            const v2f bfrag = *(const v2f*)(lptf + 4 * (t + col) + 2 * halfg);
            const v8f cz = {0.f, 0.f, 0.f, 0.f, 0.f, 0.f, 0.f, 0.f};

            v8f d = __builtin_amdgcn_wmma_f32_16x16x4_f32(
                false, afrag, false, bfrag, (short)0, cz, false, false);

            #pragma unroll
            for (int i = 0; i < 8; ++i) minD[i] = fminf(minD[i], d[i]);
        }
    }

    // ---- Finalize ----
    // Row M = v + 8*halfg lives in VGPR v across the 16 lanes of its half.
    float rowsum = 0.0f;
    #pragma unroll
    for (int v = 0; v < 8; ++v) {
        float r = minD[v];
        #pragma unroll
        for (int off = 1; off < 16; off <<= 1)
            r = fminf(r, __shfl_xor(r, off, 32));
        rowsum += r;   // lanes 0-15: rows 0..7 ; lanes 16-31: rows 8..15
    }
    rowsum += __shfl_xor(rowsum, 16, 32);   // sum of this wave's 16 row-mins

    // Re-add the dropped per-row xx: sum of xx over the wave's 16 distinct
    // rows (xx is replicated across the two half-groups).
    float xs = xx;
    #pragma unroll
    for (int off = 1; off < 16; off <<= 1) xs += __shfl_xor(xs, off, 32);

    if (lane == 0)
        partial[(size_t)blockIdx.x * WAVES_PER_BLOCK + waveId] = rowsum + xs;
}

__global__ void __launch_bounds__(256)
reduce_kernel(const float* __restrict__ partial, int n, float* __restrict__ out)
{
    __shared__ float sm[256];
    const int t = threadIdx.x;
    float s = 0.0f;
    for (int i = t; i < n; i += 256) s += partial[i];
    sm[t] = s;
    __syncthreads();
    for (int off = 128; off > 0; off >>= 1) {
        if (t < off) sm[t] += sm[t + off];
        __syncthreads();
    }
    if (t == 0) out[0] = sm[0];
}

extern "C" void kernel_launch(void* const* d_in, const int* in_sizes, int n_in,
                              void* d_out, int out_size, void* d_ws, size_t ws_size,
                              hipStream_t stream) {
    const float* preds = (const float*)d_in[0];   // [B, M, 3]
    const float* gts   = (const float*)d_in[1];   // [B, N, 3]
    const int B = 8, D = 3;
    const int M = in_sizes[0] / (B * D);          // 8192
    const int N = in_sizes[1] / (B * D);          // 8192

    float* partial = (float*)d_ws;
    const int bpbM = M / ROWS_PER_BLOCK;
    const int bpbN = N / ROWS_PER_BLOCK;
    const int waves0 = B * bpbM * WAVES_PER_BLOCK;
    const int waves1 = B * bpbN * WAVES_PER_BLOCK;

    // loss_1: for each pred point, nearest gt.  loss_2: for each gt, nearest pred.
    chamfer_nn_kernel<<<B * bpbM, 256, 0, stream>>>(preds, gts, M, N, bpbM, partial);
    chamfer_nn_kernel<<<B * bpbN, 256, 0, stream>>>(gts, preds, N, M, bpbN, partial + waves0);
    reduce_kernel<<<1, 256, 0, stream>>>(partial, waves0 + waves1, (float*)d_out);
}